// Grace_87265145520542
// MI455X (gfx1250) — compile-verified
//
#include <hip/hip_runtime.h>

typedef float v2f __attribute__((ext_vector_type(2)));
typedef float v8f __attribute__((ext_vector_type(8)));

#define NHID   64
#define KCHUNK 32
#define APAD   (KCHUNK + 4)  // stride 36 ≡ 4 (mod 32): conflict-free b64 frag reads, 16B aligned

#if defined(__gfx1250__) && __has_builtin(__builtin_amdgcn_global_load_async_to_lds_b128) && \
    __has_builtin(__builtin_amdgcn_global_load_async_to_lds_b32) && \
    __has_builtin(__builtin_amdgcn_s_wait_asynccnt)
#define USE_ASYNC_LDS 1
typedef int v4i_vs __attribute__((vector_size(16)));
typedef __attribute__((address_space(1))) v4i_vs* gptr_b128;
typedef __attribute__((address_space(3))) v4i_vs* lptr_b128;
typedef __attribute__((address_space(1))) int*     gptr_b32;
typedef __attribute__((address_space(3))) int*     lptr_b32;
#else
#define USE_ASYNC_LDS 0
#endif

// ---------------- degree / normalization ----------------

__global__ __launch_bounds__(256) void deg_init_kernel(float* deg, int n) {
  int i = blockIdx.x * 256 + threadIdx.x;
  if (i < n) deg[i] = 1.0f;  // implicit self-loop weight
}

__global__ __launch_bounds__(256) void deg_accum_kernel(float* deg, const int* __restrict__ dst,
                                                        const float* __restrict__ ew, int E) {
  int i = blockIdx.x * 256 + threadIdx.x;
  if (i < E) atomicAdd(deg + dst[i], ew[i]);
}

__global__ __launch_bounds__(256) void rsqrt_kernel(float* d, int n) {
  int i = blockIdx.x * 256 + threadIdx.x;
  if (i < n) d[i] = rsqrtf(d[i]);
}

// ---------------- tiled WMMA f32 GEMM:  C[M,64] = A[M,K] @ W[K,64] ----------------
// block = 256 threads = 8 waves; each wave owns a 16-row x 64-col output strip.
// Double-buffered LDS staging via async global->LDS DMA (ASYNCcnt) when available.

__global__ __launch_bounds__(256) void gemm_wmma_kernel(
    const float* __restrict__ A, const float* __restrict__ W,
    float* __restrict__ C, int M, int K) {
  __shared__ __align__(16) float xs[2][128 * APAD];  // A tiles: 128 rows x 32 K (padded)
  __shared__ __align__(16) float wt[2][64 * APAD];   // W chunks transposed: wt[n][kk]

  const int tid  = threadIdx.x;
  const int lane = tid & 31;
  const int wave = tid >> 5;       // 0..7
  const int half = lane >> 4;      // ISA 16x4 f32 A layout: lanes16-31 hold K+2
  const int l16  = lane & 15;
  const int r0   = blockIdx.x * 128;
  const int m0   = r0 + wave * 16;

  v8f acc0 = {}; v8f acc1 = {}; v8f acc2 = {}; v8f acc3 = {};

  auto stage = [&](int buf, int k0) {
    // A tile: 128 rows x 32 K as float4 -> 1024 x b128 transfers (4 per thread)
    for (int i = tid; i < 128 * (KCHUNK / 4); i += 256) {
      const int r  = i >> 3;       // 0..127
      const int c4 = i & 7;        // 0..7
      int row = r0 + r; if (row >= M) row = M - 1;  // clamp: tail strips never stored
      const float* gp = A + (size_t)row * K + k0 + c4 * 4;
      float* sp = &xs[buf][r * APAD + c4 * 4];
#if USE_ASYNC_LDS
      __builtin_amdgcn_global_load_async_to_lds_b128(
          (gptr_b128)(v4i_vs*)const_cast<float*>(gp),
          (lptr_b128)(v4i_vs*)sp, 0, 0);
#else
      const float4 v = *(const float4*)gp;
      if (k0 + KCHUNK < K) __builtin_prefetch(gp + KCHUNK, 0, 0);
      sp[0] = v.x; sp[1] = v.y; sp[2] = v.z; sp[3] = v.w;
#endif
    }
    // W chunk transposed (32x64): wt[n][kk] = W[(k0+kk)*64+n] -> 2048 x b32 (8 per thread)
    for (int i = tid; i < KCHUNK * 64; i += 256) {
      const int kk = i >> 6;
      const int n  = i & 63;
      const float* gp = W + (size_t)(k0 + kk) * NHID + n;
      float* sp = &wt[buf][n * APAD + kk];
#if USE_ASYNC_LDS
      __builtin_amdgcn_global_load_async_to_lds_b32(
          (gptr_b32)(int*)const_cast<float*>(gp),
          (lptr_b32)(int*)sp, 0, 0);
#else
      *sp = *gp;
#endif
    }
  };

  stage(0, 0);                         // prologue: DMA chunk 0
  const int nChunks = K / KCHUNK;
  for (int c = 0; c < nChunks; ++c) {
    const int cur = c & 1;
#if USE_ASYNC_LDS
    __builtin_amdgcn_s_wait_asynccnt(0);  // this wave's DMA into 'cur' has landed
#endif
    __syncthreads();                      // all waves' data visible; prev buffer's readers done
    if (c + 1 < nChunks) stage(cur ^ 1, (c + 1) * KCHUNK);  // overlap next DMA with compute

    const float* xr = &xs[cur][(wave * 16 + l16) * APAD];
    const float* wb = &wt[cur][0];
#pragma unroll
    for (int ks = 0; ks < KCHUNK / 4; ++ks) {
      const int kk = ks * 4 + half * 2;   // lanes0-15: K=kk,kk+1; lanes16-31: +2
      v2f a  = *(const v2f*)(xr + kk);
      v2f b0 = *(const v2f*)(wb + (l16 +  0) * APAD + kk);
      v2f b1 = *(const v2f*)(wb + (l16 + 16) * APAD + kk);
      v2f b2 = *(const v2f*)(wb + (l16 + 32) * APAD + kk);
      v2f b3 = *(const v2f*)(wb + (l16 + 48) * APAD + kk);
      acc0 = __builtin_amdgcn_wmma_f32_16x16x4_f32(false, a, false, b0, (short)0, acc0, false, false);
      acc1 = __builtin_amdgcn_wmma_f32_16x16x4_f32(false, a, false, b1, (short)0, acc1, false, false);
      acc2 = __builtin_amdgcn_wmma_f32_16x16x4_f32(false, a, false, b2, (short)0, acc2, false, false);
      acc3 = __builtin_amdgcn_wmma_f32_16x16x4_f32(false, a, false, b3, (short)0, acc3, false, false);
    }
  }

  // C/D layout: VGPR v, lanes0-15 -> M=v, lanes16-31 -> M=v+8; N = lane&15 (+16*tile)
  if (m0 < M) {  // M % 16 == 0, so strip-level guard suffices
#pragma unroll
    for (int v = 0; v < 8; ++v) {
      const int row = m0 + half * 8 + v;
      float* cp = C + (size_t)row * NHID + l16;
      cp[0]  = acc0[v];
      cp[16] = acc1[v];
      cp[32] = acc2[v];
      cp[48] = acc3[v];
    }
  }
}

// ---------------- edge scatter:  agg[dst] += h[src] * (dinv[src]*ew*dinv[dst]) ----------------
// 16 threads per edge, 4 feats each; h (12.8 MB) is L2-resident on a 192 MB L2.

__global__ __launch_bounds__(256) void scatter_kernel(
    const float* __restrict__ h, const int* __restrict__ src, const int* __restrict__ dst,
    const float* __restrict__ ew, const float* __restrict__ dinv,
    float* __restrict__ agg, int E) {
  const int gid = blockIdx.x * 256 + threadIdx.x;
  const int e = gid >> 4;
  if (e >= E) return;
  const int q = gid & 15;
  const int s = src[e];
  const int d = dst[e];
  const float c = dinv[s] * ew[e] * dinv[d];
  const float4 hv = *(const float4*)(h + (size_t)s * NHID + q * 4);
  float* ap = agg + (size_t)d * NHID + q * 4;
  atomicAdd(ap + 0, hv.x * c);
  atomicAdd(ap + 1, hv.y * c);
  atomicAdd(ap + 2, hv.z * c);
  atomicAdd(ap + 3, hv.w * c);
}

// ---------------- finalize:  out = relu(agg + h*dinv^2 + bias) ----------------

__global__ __launch_bounds__(256) void finalize_kernel(
    const float* __restrict__ agg, const float* __restrict__ h,
    const float* __restrict__ dinv, const float* __restrict__ bias,
    float* __restrict__ out, int M) {
  const int gid = blockIdx.x * 256 + threadIdx.x;
  const int nd = gid >> 4;
  if (nd >= M) return;
  const int q = gid & 15;
  const float di = dinv[nd];
  const float sl = di * di;
  const float4 a  = *(const float4*)(agg + (size_t)nd * NHID + q * 4);
  const float4 hv = *(const float4*)(h   + (size_t)nd * NHID + q * 4);
  const float4 b  = *(const float4*)(bias + q * 4);
  float4 r;
  r.x = fmaxf(a.x + hv.x * sl + b.x, 0.f);
  r.y = fmaxf(a.y + hv.y * sl + b.y, 0.f);
  r.z = fmaxf(a.z + hv.z * sl + b.z, 0.f);
  r.w = fmaxf(a.w + hv.w * sl + b.w, 0.f);
  *(float4*)(out + (size_t)nd * NHID + q * 4) = r;
}

// ---------------- host orchestration ----------------

extern "C" void kernel_launch(void* const* d_in, const int* in_sizes, int n_in,
                              void* d_out, int out_size, void* d_ws, size_t ws_size,
                              hipStream_t stream) {
  const float* x  = (const float*)d_in[0];
  const int*   ei = (const int*)  d_in[1];
  const float* ew = (const float*)d_in[2];
  const float* W1 = (const float*)d_in[3];
  const float* b1 = (const float*)d_in[4];
  const float* W2 = (const float*)d_in[5];
  const float* b2 = (const float*)d_in[6];

  const int NFEAT = 512;
  const int M = in_sizes[0] / NFEAT;   // 50000 nodes
  const int E = in_sizes[1] / 2;       // 800000 edges
  const int* src = ei;
  const int* dst = ei + E;

  float* ws = (float*)d_ws;
  const size_t nodeAlign = ((size_t)M + 63) & ~(size_t)63;
  float* dinv = ws;                          // [M]
  float* bufA = ws + nodeAlign;              // [M*64] GEMM output (h)
  float* bufB = bufA + (size_t)M * NHID;     // [M*64] aggregation accumulator
  float* bufC = bufB + (size_t)M * NHID;     // [M*64] layer-1 activations
  float* out  = (float*)d_out;

  const int nodeBlocks  = (M + 255) / 256;
  const int edgeBlocks  = (E + 255) / 256;
  const int gemmBlocks  = (M + 127) / 128;
  const int featNBlocks = (M * 16 + 255) / 256;   // node x float4 threads
  const int featEBlocks = (E * 16 + 255) / 256;   // edge x float4 threads

  // normalization: deg = 1 + segment_sum(ew, dst);  dinv = rsqrt(deg)
  deg_init_kernel <<<nodeBlocks, 256, 0, stream>>>(dinv, M);
  deg_accum_kernel<<<edgeBlocks, 256, 0, stream>>>(dinv, dst, ew, E);
  rsqrt_kernel    <<<nodeBlocks, 256, 0, stream>>>(dinv, M);

  // ---- layer 1 ----
  gemm_wmma_kernel<<<gemmBlocks, 256, 0, stream>>>(x, W1, bufA, M, NFEAT);
  (void)hipMemsetAsync(bufB, 0, sizeof(float) * (size_t)M * NHID, stream);
  scatter_kernel  <<<featEBlocks, 256, 0, stream>>>(bufA, src, dst, ew, dinv, bufB, E);
  finalize_kernel <<<featNBlocks, 256, 0, stream>>>(bufB, bufA, dinv, b1, bufC, M);

  // ---- layer 2 ----
  gemm_wmma_kernel<<<gemmBlocks, 256, 0, stream>>>(bufC, W2, bufA, M, NHID);
  (void)hipMemsetAsync(bufB, 0, sizeof(float) * (size_t)M * NHID, stream);
  scatter_kernel  <<<featEBlocks, 256, 0, stream>>>(bufA, src, dst, ew, dinv, bufB, E);
  finalize_kernel <<<featNBlocks, 256, 0, stream>>>(bufB, bufA, dinv, b2, out, M);
}